// MultiheadGEAttention_60662118089197
// MI455X (gfx1250) — compile-verified
//
#include <hip/hip_runtime.h>
#include <math.h>

// MI455X / gfx1250: wave32, WMMA (fp32 16x16x4), 320KB LDS per WGP,
// async global->LDS copies (ASYNCcnt).

typedef __attribute__((ext_vector_type(2))) float v2f;
typedef __attribute__((ext_vector_type(4))) float v4f;
typedef __attribute__((ext_vector_type(8))) float v8f;
typedef __attribute__((ext_vector_type(4))) int   v4i;

#define D_MODEL 512
#define NCOLS   128
#define NHEAD   8
#define DHEAD   64
#define DCOL    14
#define BATCH   512

// ---- async global->LDS copy (gfx1250 GLOBAL_LOAD_ASYNC_TO_LDS_B128) -------
#if defined(__has_builtin)
#if __has_builtin(__builtin_amdgcn_global_load_async_to_lds_b128)
#define HAVE_ASYNC_LDS 1
#endif
#endif

#ifdef HAVE_ASYNC_LDS
typedef __attribute__((address_space(1))) v4i* as1_v4ip;
typedef __attribute__((address_space(3))) v4i* as3_v4ip;
#endif

__device__ __forceinline__ void cp16_async(float* lds, const float* g) {
#ifdef HAVE_ASYNC_LDS
    __builtin_amdgcn_global_load_async_to_lds_b128(
        (as1_v4ip)const_cast<float*>(g), (as3_v4ip)lds, 0, 0);
#else
    *(v4f*)lds = *(const v4f*)g;
#endif
}

__device__ __forceinline__ void cp_wait() {
#ifdef HAVE_ASYNC_LDS
#if __has_builtin(__builtin_amdgcn_s_wait_asynccnt)
    __builtin_amdgcn_s_wait_asynccnt(0);
#else
    asm volatile("s_wait_asynccnt 0x0" ::: "memory");
#endif
#endif
}

__device__ __forceinline__ v8f zero8() {
    v8f z;
#pragma unroll
    for (int i = 0; i < 8; ++i) z[i] = 0.0f;
    return z;
}

// Adjacent-pair LDS load (lowers to ds_load_b64 into an aligned VGPR pair).
__device__ __forceinline__ v2f ld2(const float* p) {
    return *(const v2f*)p;
}

// A-fragment (16x4 fp32): matrix stored row-major [row][k], leading dim ld.
// lanes 0-15: M=lane, K=kbase+{0,1}; lanes 16-31: M=lane-16, K=kbase+{2,3}.
__device__ __forceinline__ v2f fragA(const float* M, int ld, int row0, int kbase,
                                     int lm, int half) {
    return ld2(M + (row0 + lm) * ld + kbase + 2 * half);
}

// B-fragment (4x16 fp32) where B[k][n] = stored[n][k] (row-major [n][k], ld).
__device__ __forceinline__ v2f fragBn(const float* M, int ld, int kbase, int n0,
                                      int lm, int half) {
    return ld2(M + (n0 + lm) * ld + kbase + 2 * half);
}

__device__ __forceinline__ v8f wmma4(v2f a, v2f b, v8f c) {
    // D = A(16x4) * B(4x16) + C, fp32
    return __builtin_amdgcn_wmma_f32_16x16x4_f32(false, a, false, b, (short)0, c,
                                                 false, false);
}

// ---------------------------------------------------------------------------
// Kernel 0: adjacency mask [H,128,128] from normalized column embeddings.
// p = sigmoid(ch_n . ct_n + bias); p>0.5 <=> logit>0 (sigmoid monotone).
// ---------------------------------------------------------------------------
__global__ void topo_mask_kernel(const float* __restrict__ col_head,
                                 const float* __restrict__ col_tail,
                                 const float* __restrict__ bias,
                                 float* __restrict__ adjmask) {
    __shared__ float sCT[NCOLS][DCOL];
    const int h = blockIdx.x;
    const int i = threadIdx.x;  // 0..127

    float v, nrm = 0.0f;
    float ct[DCOL];
#pragma unroll
    for (int c = 0; c < DCOL; ++c) {
        v = col_tail[((size_t)h * NCOLS + i) * DCOL + c];
        ct[c] = v;
        nrm += v * v;
    }
    float inv = rsqrtf(nrm);
#pragma unroll
    for (int c = 0; c < DCOL; ++c) sCT[i][c] = ct[c] * inv;

    float chv[DCOL];
    nrm = 0.0f;
#pragma unroll
    for (int c = 0; c < DCOL; ++c) {
        v = col_head[((size_t)h * NCOLS + i) * DCOL + c];
        chv[c] = v;
        nrm += v * v;
    }
    inv = rsqrtf(nrm);
#pragma unroll
    for (int c = 0; c < DCOL; ++c) chv[c] *= inv;

    __syncthreads();
    const float bi = bias[0];
    float* row = adjmask + ((size_t)h * NCOLS + i) * NCOLS;
    for (int j = 0; j < NCOLS; ++j) {
        float t = 0.0f;
#pragma unroll
        for (int c = 0; c < DCOL; ++c) t += chv[c] * sCT[j][c];
        float logit = t + bi;
        // p forced to 0 on diagonal and for edges into readout node j==0
        bool hard = (logit > 0.0f) && (i != j) && (j != 0);
        row[j] = hard ? 0.0f : -10000.0f;
    }
}

// ---------------------------------------------------------------------------
// Kernel 1: fused per-(b,h) attention.
//   FH = x_head[b] @ Wh_h^T + bh     [128,64]   (WMMA, K=512)
//   FT = x_tail[b] @ Wh_h^T + bh     (symmetric weights)
//   FV = x_tail[b] @ Wv_h^T + bv
//   S  = FH @ (FT*rel)^T / 8 + adjmask[h]
//   P  = softmax_rows(S)  -> fr_graph output
//   XO = P @ FV           -> workspace (pre-W_out activations)
// LDS: one 64KB pool reused across phases; staging via async LDS copies.
// ---------------------------------------------------------------------------
__global__ __launch_bounds__(256) void ge_attention_kernel(
    const float* __restrict__ x_head, const float* __restrict__ x_tail,
    const float* __restrict__ Wh, const float* __restrict__ bh,
    const float* __restrict__ Wv, const float* __restrict__ bv,
    const float* __restrict__ rel_emb, const float* __restrict__ adjmask,
    float* __restrict__ fr_graph, float* __restrict__ xo) {
    __shared__ float smem[16384];  // 64 KB

    const int tid  = threadIdx.x;
    const int lane = tid & 31;
    const int wave = tid >> 5;      // 0..7, each wave owns 16 rows
    const int lm   = lane & 15;
    const int half = lane >> 4;
    const int b    = blockIdx.x >> 3;  // b-major: 8 heads of same batch adjacent
    const int h    = blockIdx.x & 7;
    const int row0 = wave * 16;

    const float* xh  = x_head + (size_t)b * NCOLS * D_MODEL;
    const float* xt  = x_tail + (size_t)b * NCOLS * D_MODEL;
    const float* WhH = Wh + (size_t)h * DHEAD * D_MODEL;  // rows h*64..h*64+63
    const float* WvH = Wv + (size_t)h * DHEAD * D_MODEL;

    // ---- Phase 1: three projections, K-chunked by 32 ----------------------
    float* sXH = smem;          // [128][32] row-major (A source)
    float* sXT = smem + 4096;   // [128][32]
    float* sWH = smem + 8192;   // [64][32]  row-major [n][k] (B source)
    float* sWV = smem + 10240;  // [64][32]

    v8f accFH[4], accFT[4], accFV[4];
#pragma unroll
    for (int t = 0; t < 4; ++t) { accFH[t] = zero8(); accFT[t] = zero8(); accFV[t] = zero8(); }

    for (int kc = 0; kc < 16; ++kc) {
        const int k0 = kc * 32;
        // x tiles: 1024 float4 per matrix, async b128 global->LDS
        for (int e = tid; e < 1024; e += 256) {
            int r = e >> 3, k4 = e & 7;  // 8 float4 per 32-float row
            cp16_async(sXH + e * 4, xh + r * D_MODEL + k0 + k4 * 4);
            cp16_async(sXT + e * 4, xt + r * D_MODEL + k0 + k4 * 4);
        }
        // weight tiles kept [n][k] exactly as in global: 512 float4 each
        for (int e = tid; e < 512; e += 256) {
            int n = e >> 3, k4 = e & 7;
            cp16_async(sWH + e * 4, WhH + n * D_MODEL + k0 + k4 * 4);
            cp16_async(sWV + e * 4, WvH + n * D_MODEL + k0 + k4 * 4);
        }
        cp_wait();        // this wave's async copies complete
        __syncthreads();  // all waves' copies complete
#pragma unroll
        for (int kk = 0; kk < 32; kk += 4) {
            v2f aH = fragA(sXH, 32, row0, kk, lm, half);
            v2f aT = fragA(sXT, 32, row0, kk, lm, half);
#pragma unroll
            for (int nt = 0; nt < 4; ++nt) {
                v2f bH = fragBn(sWH, 32, kk, nt * 16, lm, half);
                v2f bV = fragBn(sWV, 32, kk, nt * 16, lm, half);
                accFH[nt] = wmma4(aH, bH, accFH[nt]);
                accFT[nt] = wmma4(aT, bH, accFT[nt]);
                accFV[nt] = wmma4(aT, bV, accFV[nt]);
            }
        }
        __syncthreads();
    }
    // biases (zeros in this config, kept for generality)
#pragma unroll
    for (int nt = 0; nt < 4; ++nt) {
        float bHh = bh[h * DHEAD + nt * 16 + lm];
        float bVh = bv[h * DHEAD + nt * 16 + lm];
#pragma unroll
        for (int r = 0; r < 8; ++r) {
            accFH[nt][r] += bHh;
            accFT[nt][r] += bHh;
            accFV[nt][r] += bVh;
        }
    }

    // ---- Stage FH (A source, [m][d]) and FT*rel ([j][d], B via fragBn) ----
    float* sFH = smem;         // [128][64]
    float* sFT = smem + 8192;  // [128][64]
#pragma unroll
    for (int nt = 0; nt < 4; ++nt) {
        float rel = rel_emb[h * DHEAD + nt * 16 + lm];
#pragma unroll
        for (int r = 0; r < 8; ++r) {
            int m = row0 + r + 8 * half;  // C/D layout: VGPR r -> rows r / r+8
            sFH[m * 64 + nt * 16 + lm] = accFH[nt][r];
            sFT[m * 64 + nt * 16 + lm] = accFT[nt][r] * rel;
        }
    }
    __syncthreads();

    // ---- Phase 2: S = FH @ FT'^T  (K = 64) -------------------------------
    v8f accS[8];
#pragma unroll
    for (int jt = 0; jt < 8; ++jt) accS[jt] = zero8();
#pragma unroll 4
    for (int kk = 0; kk < 64; kk += 4) {
        v2f a = fragA(sFH, 64, row0, kk, lm, half);
#pragma unroll
        for (int jt = 0; jt < 8; ++jt) {
            v2f bb = fragBn(sFT, 64, kk, jt * 16, lm, half);
            accS[jt] = wmma4(a, bb, accS[jt]);
        }
    }

    // scale (1/sqrt(64)) + topology mask
    const float* am = adjmask + (size_t)h * NCOLS * NCOLS;
#pragma unroll
    for (int jt = 0; jt < 8; ++jt)
#pragma unroll
        for (int r = 0; r < 8; ++r) {
            int m = row0 + r + 8 * half;
            accS[jt][r] = accS[jt][r] * 0.125f + am[m * NCOLS + jt * 16 + lm];
        }

    // row softmax: each (r, half) pair owns one row spread over 16 lanes x 8 tiles.
    // xor masks 1,2,4,8 reduce the two 16-lane halves independently (wave32).
#pragma unroll
    for (int r = 0; r < 8; ++r) {
        float mx = accS[0][r];
#pragma unroll
        for (int jt = 1; jt < 8; ++jt) mx = fmaxf(mx, accS[jt][r]);
        for (int o = 1; o < 16; o <<= 1) mx = fmaxf(mx, __shfl_xor(mx, o, 32));
        float sum = 0.0f;
#pragma unroll
        for (int jt = 0; jt < 8; ++jt) {
            float e = __expf(accS[jt][r] - mx);
            accS[jt][r] = e;
            sum += e;
        }
        for (int o = 1; o < 16; o <<= 1) sum += __shfl_xor(sum, o, 32);
        float inv = 1.0f / sum;
#pragma unroll
        for (int jt = 0; jt < 8; ++jt) accS[jt][r] *= inv;
    }

    // fr_graph output [B,H,128,128]
    float* fr = fr_graph + ((size_t)b * NHEAD + h) * NCOLS * NCOLS;
#pragma unroll
    for (int jt = 0; jt < 8; ++jt)
#pragma unroll
        for (int r = 0; r < 8; ++r) {
            int m = row0 + r + 8 * half;
            fr[(size_t)m * NCOLS + jt * 16 + lm] = accS[jt][r];
        }

    // ---- Phase 3: XO = P @ FV, P streamed through LDS in two 64-col chunks.
    // FV stored transposed [d][j] so B-fragments (B[k=j][n=d]) are adjacent
    // pairs in LDS (b64 loads); transpose cost paid once on the store side.
    float* sP   = smem;         // [128][64] (j-chunk, A source)
    float* sFVt = smem + 8192;  // [64][128] = FV^T
    __syncthreads();            // all reads of sFH/sFT complete
#pragma unroll
    for (int nt = 0; nt < 4; ++nt)
#pragma unroll
        for (int r = 0; r < 8; ++r) {
            int j = row0 + r + 8 * half;  // FV rows are token index j
            sFVt[(nt * 16 + lm) * 128 + j] = accFV[nt][r];
        }

    v8f accXO[4];
#pragma unroll
    for (int t = 0; t < 4; ++t) accXO[t] = zero8();

    for (int c = 0; c < 2; ++c) {
#pragma unroll
        for (int jt = 0; jt < 4; ++jt)
#pragma unroll
            for (int r = 0; r < 8; ++r) {
                int m = row0 + r + 8 * half;
                sP[m * 64 + jt * 16 + lm] = accS[c * 4 + jt][r];
            }
        __syncthreads();
#pragma unroll 4
        for (int kk = 0; kk < 64; kk += 4) {
            v2f a = fragA(sP, 64, row0, kk, lm, half);
#pragma unroll
            for (int nt = 0; nt < 4; ++nt) {
                v2f bb = fragBn(sFVt, 128, c * 64 + kk, nt * 16, lm, half);
                accXO[nt] = wmma4(a, bb, accXO[nt]);
            }
        }
        __syncthreads();
    }

    // XO -> workspace, heads merged: [B,128,512], head h at cols h*64..
    float* xob = xo + (size_t)b * NCOLS * D_MODEL;
#pragma unroll
    for (int nt = 0; nt < 4; ++nt)
#pragma unroll
        for (int r = 0; r < 8; ++r) {
            int m = row0 + r + 8 * half;
            xob[(size_t)m * D_MODEL + h * DHEAD + nt * 16 + lm] = accXO[nt][r];
        }
}

// ---------------------------------------------------------------------------
// Kernel 2: out = XO @ W_out^T + b_out.  [65536x512]x[512x512]
// 128x128 output tile per workgroup, K-chunked by 32. W tile kept [n][k].
// ---------------------------------------------------------------------------
__global__ __launch_bounds__(256) void out_proj_kernel(
    const float* __restrict__ xo, const float* __restrict__ Wout,
    const float* __restrict__ bout, float* __restrict__ out) {
    __shared__ float smem[8192];  // 32KB: sA[128][32] + sB[128][32] ([n][k])
    float* sA = smem;
    float* sB = smem + 4096;

    const int tid  = threadIdx.x;
    const int lane = tid & 31;
    const int wave = tid >> 5;
    const int lm   = lane & 15;
    const int half = lane >> 4;
    const int rb   = blockIdx.x >> 2;  // 512 row blocks of 128
    const int cb   = blockIdx.x & 3;   // 4 col blocks of 128
    const size_t R0 = (size_t)rb * 128;
    const int C0   = cb * 128;
    const int row0 = wave * 16;

    v8f acc[8];
#pragma unroll
    for (int t = 0; t < 8; ++t) acc[t] = zero8();

    for (int kc = 0; kc < 16; ++kc) {
        const int k0 = kc * 32;
        for (int e = tid; e < 1024; e += 256) {
            int r = e >> 3, k4 = e & 7;
            cp16_async(sA + e * 4, xo + (R0 + r) * D_MODEL + k0 + k4 * 4);
            cp16_async(sB + e * 4, Wout + (size_t)(C0 + r) * D_MODEL + k0 + k4 * 4);
        }
        cp_wait();
        __syncthreads();
#pragma unroll
        for (int kk = 0; kk < 32; kk += 4) {
            v2f a = fragA(sA, 32, row0, kk, lm, half);
#pragma unroll
            for (int nt = 0; nt < 8; ++nt) {
                v2f bb = fragBn(sB, 32, kk, nt * 16, lm, half);
                acc[nt] = wmma4(a, bb, acc[nt]);
            }
        }
        __syncthreads();
    }
#pragma unroll
    for (int nt = 0; nt < 8; ++nt) {
        float bo = bout[C0 + nt * 16 + lm];
#pragma unroll
        for (int r = 0; r < 8; ++r) {
            int m = row0 + r + 8 * half;
            out[(R0 + m) * D_MODEL + C0 + nt * 16 + lm] = acc[nt][r] + bo;
        }
    }
}

// ---------------------------------------------------------------------------
extern "C" void kernel_launch(void* const* d_in, const int* in_sizes, int n_in,
                              void* d_out, int out_size, void* d_ws, size_t ws_size,
                              hipStream_t stream) {
    const float* x_head = (const float*)d_in[0];
    const float* x_tail = (const float*)d_in[1];
    const float* Whw    = (const float*)d_in[2];
    const float* Whb    = (const float*)d_in[3];
    const float* Wvw    = (const float*)d_in[4];
    const float* Wvb    = (const float*)d_in[5];
    const float* Wow    = (const float*)d_in[6];
    const float* Wob    = (const float*)d_in[7];
    const float* rel    = (const float*)d_in[8];
    const float* ch     = (const float*)d_in[9];
    const float* ct     = (const float*)d_in[10];
    const float* bias   = (const float*)d_in[11];

    float* out_x  = (float*)d_out;                                   // [B,128,512]
    float* out_fr = out_x + (size_t)BATCH * NCOLS * D_MODEL;         // [B,H,128,128]

    float* adjmask = (float*)d_ws;                                   // 512 KB
    float* xo = (float*)((char*)d_ws +
                         (size_t)NHEAD * NCOLS * NCOLS * sizeof(float));  // 128 MB

    topo_mask_kernel<<<NHEAD, NCOLS, 0, stream>>>(ch, ct, bias, adjmask);
    ge_attention_kernel<<<BATCH * NHEAD, 256, 0, stream>>>(
        x_head, x_tail, Whw, Whb, Wvw, Wvb, rel, adjmask, out_fr, xo);
    out_proj_kernel<<<(65536 / 128) * (D_MODEL / 128), 256, 0, stream>>>(
        xo, Wow, Wob, out_x);
}